// ScoreEntropyDiscreteDiffusino_20581483282432
// MI455X (gfx1250) — compile-verified
//
#include <hip/hip_runtime.h>
#include <cstdint>
#include <cmath>

// ScoreEntropyDiscreteDiffusion q_sample, fused single pass for MI455X (gfx1250).
//
// Key algebraic collapse: Q is uniform (1/K * ones), so the t-step scan reduces
// to a two-scalar recurrence (q_hot, q_other) identical for all (b,f). The
// kernel is then a pure streaming op: 256 MB in, 128 MB out -> HBM-bound
// (~17 us at 23.3 TB/s). Data movement uses the CDNA5 async global->LDS path
// (global_load_async_to_lds_b128 + s_wait_asynccnt double buffering); argmax
// over K=32 bins is a wave32 shfl_xor butterfly over 8 lanes x float4.

typedef __attribute__((ext_vector_type(4))) float v4f;

#define B_DIM    8192
#define F_DIM    128
#define K_DIM    32
#define N_TOT    (B_DIM * F_DIM * K_DIM)   // 33,554,432 elements
#define T_SCHED  1000
#define NBLOCKS  2048
#define NTHREADS 256
#define CHUNK    (NTHREADS * 4)            // 1024 elements per chunk
#define NCHUNKS  (N_TOT / CHUNK)           // 32768
#define CPB      (NCHUNKS / NBLOCKS)       // 16 chunks per block (exact)

__global__ __launch_bounds__(NTHREADS) void sedd_qsample_kernel(
    const float* __restrict__ log_x0,
    const float* __restrict__ u,
    const int* __restrict__ t_ptr,
    float* __restrict__ out)
{
    __shared__ v4f sx[2][NTHREADS];
    __shared__ v4f su[2][NTHREADS];

    const int tid = (int)threadIdx.x;
    const int t   = *t_ptr;

    // ---- Closed-form collapse of the t-step scan (two scalars) ----
    // q_{k+1} = (1-sigma_k) q_k + sigma_k * S_k / 32 + 1e-8,  S = q_hot + 31*q_other
    // sigma_k from the cosine schedule; final log_p values are two constants.
    float qh = 1.0f + 1e-8f;   // == 1.0f in f32, matching jnp one_hot + eps
    float qo = 1e-8f;
    const float HALF_PI = 1.57079632679489662f;
    float c0 = cosf((0.008f / 1.008f) * HALF_PI);
    float ac_prev = c0 * c0;
    for (int k = 1; k <= t; ++k) {
        float a  = ((float)k / (float)T_SCHED + 0.008f) / 1.008f * HALF_PI;
        float ck = cosf(a);
        float ac = ck * ck;
        float sigma = 1.0f - ac / ac_prev;
        ac_prev = ac;
        float S = qh + 31.0f * qo;
        float m = sigma * (S * (1.0f / 32.0f));
        qh = (1.0f - sigma) * qh + m + 1e-8f;
        qo = (1.0f - sigma) * qo + m + 1e-8f;
    }
    const float Lhot = logf(qh);
    const float Loth = logf(qo);
    const float OUT_HOT = logf(1.0f + 1e-8f);   // folds to 0.0f
    const float OUT_OTH = logf(1e-8f);          // folds to -18.420681f

    // LDS byte offsets for this thread's double-buffer slots (low 32 bits of
    // the generic address == LDS offset on gfx1250 flat aperture).
    const uint32_t dlx[2] = { (uint32_t)(uintptr_t)&sx[0][tid],
                              (uint32_t)(uintptr_t)&sx[1][tid] };
    const uint32_t dlu[2] = { (uint32_t)(uintptr_t)&su[0][tid],
                              (uint32_t)(uintptr_t)&su[1][tid] };

    const long base_chunk = (long)blockIdx.x * CPB;

    // Issue one chunk's async loads (B128 per thread per array) into buffer b.
    auto issue = [&](int b, long cidx) {
        const uint64_t gx = (uint64_t)(uintptr_t)(log_x0 + cidx * CHUNK + tid * 4);
        const uint64_t gu = (uint64_t)(uintptr_t)(u      + cidx * CHUNK + tid * 4);
        asm volatile("global_load_async_to_lds_b128 %0, %1, off"
                     :: "v"(dlx[b]), "v"(gx) : "memory");
        asm volatile("global_load_async_to_lds_b128 %0, %1, off"
                     :: "v"(dlu[b]), "v"(gu) : "memory");
    };

    issue(0, base_chunk);   // prime the pipeline

    for (int i = 0; i < CPB; ++i) {
        const int cur = i & 1;
        if (i + 1 < CPB) {
            issue(cur ^ 1, base_chunk + i + 1);
            // Keep the 2 just-issued loads in flight; async loads complete in
            // order, so <=2 outstanding means chunk i's data is in LDS.
            asm volatile("s_wait_asynccnt 0x2" ::: "memory");
        } else {
            asm volatile("s_wait_asynccnt 0x0" ::: "memory");
        }

        const v4f lx = sx[cur][tid];   // ds_load_b128 (own async-written slot)
        const v4f lu = su[cur][tid];

        // score = log_p + gumbel(u); log_p is a 2-way select on the one-hot.
        const int bin0 = (tid & 7) * 4;
        float g0 = -logf(-logf(lu[0] + 1e-30f) + 1e-30f);
        float g1 = -logf(-logf(lu[1] + 1e-30f) + 1e-30f);
        float g2 = -logf(-logf(lu[2] + 1e-30f) + 1e-30f);
        float g3 = -logf(-logf(lu[3] + 1e-30f) + 1e-30f);
        float s0 = ((lx[0] > -9.0f) ? Lhot : Loth) + g0;
        float s1 = ((lx[1] > -9.0f) ? Lhot : Loth) + g1;
        float s2 = ((lx[2] > -9.0f) ? Lhot : Loth) + g2;
        float s3 = ((lx[3] > -9.0f) ? Lhot : Loth) + g3;

        // Local argmax over 4 bins (lowest index wins ties, like jnp.argmax).
        float bv = s0; int bi = bin0;
        if (s1 > bv) { bv = s1; bi = bin0 + 1; }
        if (s2 > bv) { bv = s2; bi = bin0 + 2; }
        if (s3 > bv) { bv = s3; bi = bin0 + 3; }

        // Butterfly lex-max over the 8 lanes of this feature (wave32).
        #pragma unroll
        for (int off = 4; off >= 1; off >>= 1) {
            float ov = __shfl_xor(bv, off, 8);
            int   oi = __shfl_xor(bi, off, 8);
            if (ov > bv || (ov == bv && oi < bi)) { bv = ov; bi = oi; }
        }

        v4f o;
        o[0] = (bi == bin0 + 0) ? OUT_HOT : OUT_OTH;
        o[1] = (bi == bin0 + 1) ? OUT_HOT : OUT_OTH;
        o[2] = (bi == bin0 + 2) ? OUT_HOT : OUT_OTH;
        o[3] = (bi == bin0 + 3) ? OUT_HOT : OUT_OTH;

        // Output is stream-once: non-temporal b128 store, keep L2 for inputs.
        v4f* dst = (v4f*)(out + (base_chunk + i) * (long)CHUNK + tid * 4);
        __builtin_nontemporal_store(o, dst);
    }
}

extern "C" void kernel_launch(void* const* d_in, const int* in_sizes, int n_in,
                              void* d_out, int out_size, void* d_ws, size_t ws_size,
                              hipStream_t stream) {
    (void)in_sizes; (void)n_in; (void)out_size; (void)d_ws; (void)ws_size;
    const float* log_x0 = (const float*)d_in[0];
    const float* uu     = (const float*)d_in[1];
    const int*   t      = (const int*)d_in[2];
    float*       out    = (float*)d_out;
    sedd_qsample_kernel<<<NBLOCKS, NTHREADS, 0, stream>>>(log_x0, uu, t, out);
}